// SparseDistributedRepresentation_81226421502241
// MI455X (gfx1250) — compile-verified
//
#include <hip/hip_runtime.h>
#include <hip/hip_bf16.h>

typedef __attribute__((ext_vector_type(2))) float v2f;
typedef __attribute__((ext_vector_type(8))) float v8f;

#define BATCH   16384
#define IN_SIZE 512
#define REP     1024
#define TOPK    51

// ---------------------------------------------------------------------------
// Tiled transpose (32x32 tiles, 256 threads): out[C][R] = in[R][C].
// zerodiag: zero elements where r==c (lateral-inhibition mask).
// ---------------------------------------------------------------------------
__global__ __launch_bounds__(256) void transpose_kernel(const float* __restrict__ in,
                                                        float* __restrict__ out,
                                                        int R, int C, int zerodiag) {
    __shared__ float tile[32][33];
    const int tx = threadIdx.x & 31;
    const int tyb = threadIdx.x >> 5;            // 0..7
    const int r0 = blockIdx.y * 32;
    const int c0 = blockIdx.x * 32;
#pragma unroll
    for (int i = 0; i < 4; ++i) {
        const int ty = tyb + i * 8;
        tile[ty][tx] = in[(size_t)(r0 + ty) * C + c0 + tx];
    }
    __syncthreads();
#pragma unroll
    for (int i = 0; i < 4; ++i) {
        const int ty = tyb + i * 8;
        const int oc = r0 + tx;                  // original row
        const int orow = c0 + ty;                // original col
        float v = tile[tx][ty];                  // = in[oc][orow]
        if (zerodiag && orow == oc) v = 0.0f;
        out[(size_t)orow * R + oc] = v;
    }
}

__global__ __launch_bounds__(256) void zero_accum_kernel(float* __restrict__ colsum,
                                                         float* __restrict__ poscnt) {
    int i = blockIdx.x * 256 + threadIdx.x;
    if (i < REP) colsum[i] = 0.0f;
    if (i == 0)  poscnt[0] = 0.0f;
}

// ---------------------------------------------------------------------------
// Async global -> LDS staging (CDNA5): per-lane 16B copy, tracked by ASYNCcnt.
// lds_off = low 32 bits of a generic pointer to __shared__ (aperture form is
// {SHARED_BASE, lds_byte_offset}; hardware adds LDS_BASE).
// ---------------------------------------------------------------------------
__device__ __forceinline__ void async_copy_b128(unsigned lds_off, const float* gptr) {
    asm volatile("global_load_async_to_lds_b128 %0, %1, off"
                 :
                 : "v"(lds_off), "v"((unsigned long long)(const void*)gptr)
                 : "memory");
}

// ---------------------------------------------------------------------------
// WMMA f32 GEMM with BOTH operands K-major:  C[M,N] = A[M,K] @ BT[N,K]^T
//   MODE 0: out = C                        (raw projection)
//   MODE 1: out = relu(raw - C - th[col])  (lateral-inhibition iteration)
//
// Block: 256 threads (8 waves). Block tile 128(M) x 128(N), K-chunk 32,
// double-buffered LDS fed by global_load_async_to_lds_b128.
// Wave w: wm = w&3 -> 32-row strip, wn = w>>2 -> 64-col strip.
// Each wave: 2 A-frags x 4 B-frags -> 8 v8f accumulators, 8 WMMA / K-step.
//
// Fragment layout per ISA 7.12.2 (f32 16x16x4): lanes 0-15 hold K={k0,k0+1},
// lanes 16-31 hold K={k0+2,k0+3}; with K-major LDS tiles every fragment is
// one aligned ds_load_b64. Padded stride 36 floats -> 16B-aligned rows and
// conflict-free column access (gcd-free bank walk).
// ---------------------------------------------------------------------------
template <int MODE>
__global__ __launch_bounds__(256) void gemm_kernel(const float* __restrict__ A,
                                                   const float* __restrict__ BT,
                                                   const float* __restrict__ raw,
                                                   const float* __restrict__ th,
                                                   float* __restrict__ out,
                                                   int M, int N, int K) {
    __shared__ float As[2][128][36];   // 2 x 18KB
    __shared__ float Bs[2][128][36];   // 2 x 18KB  (Bs[buf][col][k])

    const int tid  = threadIdx.x;
    const int lane = tid & 31;
    const int wave = tid >> 5;
    const int wm   = wave & 3;         // 32-row strip
    const int wn   = wave >> 2;        // 64-col strip
    const int lm   = lane & 15;
    const int lh   = lane >> 4;        // selects K pair within fragment
    const int m0   = blockIdx.y * 128;
    const int n0   = blockIdx.x * 128;

    // staging coordinates: 4 x b128 per thread per tile (128 rows x 32 K)
    const int sr = tid >> 3;           // 0..31 base row (stride 32 over i)
    const int sc = (tid & 7) << 2;     // K offset 0,4,..,28
    const float* agp = A  + (size_t)(m0 + sr) * K + sc;
    const float* bgp = BT + (size_t)(n0 + sr) * K + sc;
    const unsigned a_lds0 = (unsigned)(size_t)(void*)&As[0][sr][sc];
    const unsigned a_lds1 = (unsigned)(size_t)(void*)&As[1][sr][sc];
    const unsigned b_lds0 = (unsigned)(size_t)(void*)&Bs[0][sr][sc];
    const unsigned b_lds1 = (unsigned)(size_t)(void*)&Bs[1][sr][sc];
    const unsigned rstep  = 32u * 36u * 4u;      // 32 rows of padded stride

    v8f acc[2][4] = {};

    const int nchunks = K >> 5;

    // prologue: stage chunk 0 into buffer 0
#pragma unroll
    for (int i = 0; i < 4; ++i) {
        async_copy_b128(a_lds0 + i * rstep, agp + (size_t)(i * 32) * K);
        async_copy_b128(b_lds0 + i * rstep, bgp + (size_t)(i * 32) * K);
    }

    for (int ic = 0; ic < nchunks; ++ic) {
        const int cur = ic & 1;
        if (ic + 1 < nchunks) {
            const int kn = (ic + 1) << 5;
            const unsigned al = cur ? a_lds0 : a_lds1;
            const unsigned bl = cur ? b_lds0 : b_lds1;
#pragma unroll
            for (int i = 0; i < 4; ++i) {
                async_copy_b128(al + i * rstep, agp + (size_t)(i * 32) * K + kn);
                async_copy_b128(bl + i * rstep, bgp + (size_t)(i * 32) * K + kn);
            }
            asm volatile("s_wait_asynccnt 0x8" ::: "memory");  // chunk ic landed
        } else {
            asm volatile("s_wait_asynccnt 0x0" ::: "memory");
        }
        __syncthreads();   // all waves' async data visible

#pragma unroll
        for (int kk = 0; kk < 32; kk += 4) {
            const int ka = kk + 2 * lh;
            float2 af0 = *(const float2*)&As[cur][wm * 32 + lm][ka];
            float2 af1 = *(const float2*)&As[cur][wm * 32 + 16 + lm][ka];
            v2f a0; a0[0] = af0.x; a0[1] = af0.y;
            v2f a1; a1[0] = af1.x; a1[1] = af1.y;
#pragma unroll
            for (int t = 0; t < 4; ++t) {
                float2 bf = *(const float2*)&Bs[cur][wn * 64 + t * 16 + lm][ka];
                v2f b; b[0] = bf.x; b[1] = bf.y;
                acc[0][t] = __builtin_amdgcn_wmma_f32_16x16x4_f32(
                    false, a0, false, b, (short)0, acc[0][t], false, false);
                acc[1][t] = __builtin_amdgcn_wmma_f32_16x16x4_f32(
                    false, a1, false, b, (short)0, acc[1][t], false, false);
            }
        }
        __syncthreads();   // done reading buffer before it is re-staged
    }

    // ---- epilogue: C/D layout vgpr j -> row j (+8 for upper lane half) ----
#pragma unroll
    for (int at = 0; at < 2; ++at) {
#pragma unroll
        for (int t = 0; t < 4; ++t) {
#pragma unroll
            for (int j = 0; j < 8; ++j) {
                const int row = m0 + wm * 32 + at * 16 + j + 8 * lh;
                const int col = n0 + wn * 64 + t * 16 + lm;
                const size_t idx = (size_t)row * N + col;
                float v = acc[at][t][j];
                if (MODE == 1) {
                    v = raw[idx] - v - th[col];
                    v = v > 0.0f ? v : 0.0f;
                }
                out[idx] = v;
            }
        }
    }
}

// ---------------------------------------------------------------------------
// Per-row exact top-51 via MSB radix-select on float bit patterns.
// acts >= 0 (post-ReLU) so uint order == float order; bit 31 always 0.
// ---------------------------------------------------------------------------
__global__ __launch_bounds__(256) void topk_kernel(const float* __restrict__ acts,
                                                   float* __restrict__ sparse,
                                                   float* __restrict__ colsum,
                                                   float* __restrict__ poscnt) {
    const int row = blockIdx.x;
    const int tid = threadIdx.x;
    const float* arow = acts + (size_t)row * REP;

    float    v[4];
    unsigned u[4];
#pragma unroll
    for (int i = 0; i < 4; ++i) {
        v[i] = arow[tid + i * 256];
        u[i] = __float_as_uint(v[i]);
    }

    __shared__ unsigned cnts[31];
    if (tid < 31) cnts[tid] = 0u;
    __syncthreads();

    unsigned prefix = 0u;
    unsigned kk = TOPK;
    for (int bit = 30; bit >= 0; --bit) {
        const unsigned test = prefix | (1u << bit);
        const unsigned mask = ~((1u << bit) - 1u);
        unsigned local = 0;
#pragma unroll
        for (int i = 0; i < 4; ++i) local += ((u[i] & mask) == test) ? 1u : 0u;
        if (local) atomicAdd(&cnts[30 - bit], local);
        __syncthreads();
        const unsigned c = cnts[30 - bit];
        if (c >= kk) prefix = test; else kk -= c;
    }
    // prefix = bit pattern of the 51st-largest element of this row.

    __shared__ unsigned blockPos;
    if (tid == 0) blockPos = 0u;
    __syncthreads();

    unsigned localPos = 0;
#pragma unroll
    for (int i = 0; i < 4; ++i) {
        const int col  = tid + i * 256;
        const bool keep = (u[i] != 0u) && (u[i] >= prefix);
        const float o   = keep ? v[i] : 0.0f;
        sparse[(size_t)row * REP + col] = o;
        if (keep) {
            atomicAdd(&colsum[col], o);
            ++localPos;
        }
    }
    if (localPos) atomicAdd(&blockPos, localPos);
    __syncthreads();
    if (tid == 0) atomicAdd(poscnt, (float)blockPos);
}

// ---------------------------------------------------------------------------
// finalize: threshold adaptation + actual sparsity
// ---------------------------------------------------------------------------
__global__ __launch_bounds__(256) void finalize_kernel(const float* __restrict__ th,
                                                       const float* __restrict__ colsum,
                                                       const float* __restrict__ poscnt,
                                                       float* __restrict__ out_th,
                                                       float* __restrict__ out_sp) {
    const int n = blockIdx.x * 256 + threadIdx.x;
    if (n < REP) {
        const float cur    = colsum[n] * (1.0f / (float)BATCH);   // mean over batch
        const float recent = cur * (1.0f / 100.0f);               // / HISTORY_LEN
        const float upd    = 0.01f * (recent - 0.05f) * 0.001f;   // rate*(x-target)*dt
        float nt = th[n] + upd;
        nt = fminf(fmaxf(nt, 0.01f), 2.0f);
        out_th[n] = nt;
    }
    if (n == 0) out_sp[0] = poscnt[0] * (1.0f / ((float)BATCH * (float)REP));
}

// ---------------------------------------------------------------------------
extern "C" void kernel_launch(void* const* d_in, const int* in_sizes, int n_in,
                              void* d_out, int out_size, void* d_ws, size_t ws_size,
                              hipStream_t stream) {
    const float* input = (const float*)d_in[0];   // [BATCH, IN_SIZE]
    const float* enc   = (const float*)d_in[1];   // [IN_SIZE, REP]
    const float* lat   = (const float*)d_in[2];   // [REP, REP]
    const float* th    = (const float*)d_in[3];   // [REP]
    float* out = (float*)d_out;                   // sparse | new_th | sparsity

    float* ws     = (float*)d_ws;
    float* raw    = ws;                                        // BATCH*REP
    float* actsA  = ws + (size_t)BATCH * REP;                  // BATCH*REP
    float* wlatT  = ws + (size_t)2 * BATCH * REP;              // REP*REP  (transposed, zero diag)
    float* encT   = wlatT + (size_t)REP * REP;                 // REP*IN_SIZE (transposed)
    float* colsum = encT + (size_t)REP * IN_SIZE;              // REP
    float* poscnt = colsum + REP;                              // 1
    float* actsB  = out;   // reuse d_out's sparse region as ping buffer

    transpose_kernel<<<dim3(REP / 32, IN_SIZE / 32), 256, 0, stream>>>(enc, encT,
                                                                       IN_SIZE, REP, 0);
    transpose_kernel<<<dim3(REP / 32, REP / 32), 256, 0, stream>>>(lat, wlatT,
                                                                   REP, REP, 1);
    zero_accum_kernel<<<5, 256, 0, stream>>>(colsum, poscnt);

    dim3 grid(REP / 128, BATCH / 128);
    // raw = input @ enc
    gemm_kernel<0><<<grid, 256, 0, stream>>>(input, encT, nullptr, nullptr, raw,
                                             BATCH, REP, IN_SIZE);
    // 5 lateral-inhibition iterations (ping-pong; odd count ends in actsA)
    gemm_kernel<1><<<grid, 256, 0, stream>>>(raw,   wlatT, raw, th, actsA, BATCH, REP, REP);
    gemm_kernel<1><<<grid, 256, 0, stream>>>(actsA, wlatT, raw, th, actsB, BATCH, REP, REP);
    gemm_kernel<1><<<grid, 256, 0, stream>>>(actsB, wlatT, raw, th, actsA, BATCH, REP, REP);
    gemm_kernel<1><<<grid, 256, 0, stream>>>(actsA, wlatT, raw, th, actsB, BATCH, REP, REP);
    gemm_kernel<1><<<grid, 256, 0, stream>>>(actsB, wlatT, raw, th, actsA, BATCH, REP, REP);

    topk_kernel<<<BATCH, 256, 0, stream>>>(actsA, out, colsum, poscnt);

    finalize_kernel<<<4, 256, 0, stream>>>(th, colsum, poscnt,
                                           out + (size_t)BATCH * REP,
                                           out + (size_t)BATCH * REP + REP);
}